// GeoDecoder_67147518705770
// MI455X (gfx1250) — compile-verified
//
#include <hip/hip_runtime.h>
#include <math.h>

// ---------------------------------------------------------------------------
// GeoDecoder feature propagation for MI455X (gfx1250, wave32, WMMA).
//
// Per stage: dist[n,s] = ||xyzf_n - xyzc_s|| + sigmoid(||nrmf_n - nrmc_s||),
// top-8 smallest per n, inverse-distance weights, gather C=256 coarse feats,
// merge with fine feats:  (max(p1,interp) + (p1+interp)*0.5)*0.5.
//
// Distance tiles are computed with exact-f32 V_WMMA_F32_16X16X4_F32:
//   A row  (coarse m) = ( cx, cy, cz, |c|^2 )
//   B col  (fine   n) = (-2fx,-2fy,-2fz, 1  )
//   C[m][n]           = |f_n|^2
//   => D[m][n] = |c_m - f_n|^2   (full f32 precision, RNE)
//
// Scoring uses the hardware TRANS unit directly (v_sqrt_f32 / v_rcp_f32,
// ~1 ulp) instead of the IEEE-exact software expansions: inputs are O(1)
// and the result only feeds a top-k select + inverse-distance weights.
// ---------------------------------------------------------------------------

typedef __attribute__((ext_vector_type(2))) float v2f;
typedef __attribute__((ext_vector_type(8))) float v8f;

#define KNN   8
#define CCH   256      // feature channels
#define WAVE  32
#define WPB   8        // waves per block
#define BLOCK (WAVE * WPB)
#define TPW   16       // fine points per wave (WMMA N dimension)

__device__ __forceinline__ float fast_sqrt(float x) {
    return __builtin_amdgcn_sqrtf(x);        // raw v_sqrt_f32 (TRANS unit)
}
__device__ __forceinline__ float fast_rcp(float x) {
    return __builtin_amdgcn_rcpf(x);         // raw v_rcp_f32 (TRANS unit)
}
__device__ __forceinline__ float sigmoid_f(float x) {
    return fast_rcp(1.0f + __expf(-x));
}

// Insert candidate (v,s) into sorted-ascending top-8 kept in registers.
// Single guarded bubble pass (array is sorted before the insert).
__device__ __forceinline__ void topk_insert(float (&bd)[KNN], int (&bi)[KNN],
                                            float v, int s) {
    if (v < bd[KNN - 1]) {
        bd[KNN - 1] = v;
        bi[KNN - 1] = s;
#pragma unroll
        for (int j = KNN - 1; j > 0; --j) {
            if (bd[j] < bd[j - 1]) {
                float td = bd[j]; bd[j] = bd[j - 1]; bd[j - 1] = td;
                int   ti = bi[j]; bi[j] = bi[j - 1]; bi[j - 1] = ti;
            }
        }
    }
}

__global__ __launch_bounds__(BLOCK)
void propagate_kernel(const float* __restrict__ xyzf,   // [B,N,3] fine coords
                      const float* __restrict__ xyzc,   // [B,S,3] coarse coords
                      const float* __restrict__ nrmf,   // [B,N,3]
                      const float* __restrict__ nrmc,   // [B,S,3]
                      const float* __restrict__ pf,     // [B,N,C] fine feats
                      const float* __restrict__ pc,     // [B,S,C] coarse feats
                      float*       __restrict__ out,    // [B,N,C]
                      int N, int S)
{
    // Per-wave scratch: raw half-lane candidates, then final (idx, weight).
    __shared__ float s_d [WPB][WAVE][KNN];
    __shared__ int   s_i [WPB][WAVE][KNN];
    __shared__ float s_w [WPB][TPW][KNN];
    __shared__ int   s_fi[WPB][TPW][KNN];

    const int lane = threadIdx.x & (WAVE - 1);
    const int wv   = threadIdx.x >> 5;
    const int half = lane >> 4;        // which K-half / M-half this lane feeds
    const int m16  = lane & 15;

    const int tile          = blockIdx.x * WPB + wv;   // 16 fine points / tile
    const int tilesPerBatch = N / TPW;
    const int b             = tile / tilesPerBatch;
    const int n0            = (tile % tilesPerBatch) * TPW;

    // ---- B matrix + C accumulator from this lane's fine point (fixed) ------
    const int   nf  = n0 + m16;
    const float fx  = xyzf[((size_t)b * N + nf) * 3 + 0];
    const float fy  = xyzf[((size_t)b * N + nf) * 3 + 1];
    const float fz  = xyzf[((size_t)b * N + nf) * 3 + 2];
    const float gx  = nrmf[((size_t)b * N + nf) * 3 + 0];
    const float gy  = nrmf[((size_t)b * N + nf) * 3 + 1];
    const float gz  = nrmf[((size_t)b * N + nf) * 3 + 2];
    const float fsq = fx * fx + fy * fy + fz * fz;
    const float gsq = gx * gx + gy * gy + gz * gz;

    v2f Bx, Bn;
    Bx.x = half ? (-2.0f * fz) : (-2.0f * fx);   // K=2 : K=0
    Bx.y = half ? 1.0f         : (-2.0f * fy);   // K=3 : K=1
    Bn.x = half ? (-2.0f * gz) : (-2.0f * gx);
    Bn.y = half ? 1.0f         : (-2.0f * gy);

    v8f Cx, Cn;
#pragma unroll
    for (int r = 0; r < 8; ++r) { Cx[r] = fsq; Cn[r] = gsq; }

    // ---- sweep coarse points in 16-wide chunks, exact-f32 WMMA -------------
    float bd[KNN]; int bi[KNN];
#pragma unroll
    for (int k = 0; k < KNN; ++k) { bd[k] = 3.0e38f; bi[k] = 0; }

    for (int s0 = 0; s0 < S; s0 += 16) {
        const int   sc  = s0 + m16;                  // A row this lane feeds
        const float cx  = xyzc[((size_t)b * S + sc) * 3 + 0];
        const float cy  = xyzc[((size_t)b * S + sc) * 3 + 1];
        const float cz  = xyzc[((size_t)b * S + sc) * 3 + 2];
        const float hx  = nrmc[((size_t)b * S + sc) * 3 + 0];
        const float hy  = nrmc[((size_t)b * S + sc) * 3 + 1];
        const float hz  = nrmc[((size_t)b * S + sc) * 3 + 2];
        const float csq = cx * cx + cy * cy + cz * cz;
        const float hsq = hx * hx + hy * hy + hz * hz;

        v2f Ax, An;
        Ax.x = half ? cz  : cx;   // K=2 : K=0
        Ax.y = half ? csq : cy;   // K=3 : K=1
        An.x = half ? hz  : hx;
        An.y = half ? hsq : hy;

        // D[m][n] = squared distance tile (16 coarse x 16 fine), exact f32.
        v8f Dx = __builtin_amdgcn_wmma_f32_16x16x4_f32(
            false, Ax, false, Bx, (short)0, Cx, false, false);
        v8f Dn = __builtin_amdgcn_wmma_f32_16x16x4_f32(
            false, An, false, Bn, (short)0, Cn, false, false);

        // Lane holds rows m = r + 8*half of its fine-point column.
#pragma unroll
        for (int r = 0; r < 8; ++r) {
            float dist = fast_sqrt(fmaxf(Dx[r], 0.0f)) +
                         sigmoid_f(fast_sqrt(fmaxf(Dn[r], 0.0f)));
            topk_insert(bd, bi, dist, s0 + r + 8 * half);
        }
    }

    // ---- merge the two half-lane candidate lists through LDS ---------------
#pragma unroll
    for (int k = 0; k < KNN; ++k) { s_d[wv][lane][k] = bd[k]; s_i[wv][lane][k] = bi[k]; }
    __syncthreads();

    if (lane < TPW) {
        float fd[KNN]; int fi[KNN];
#pragma unroll
        for (int k = 0; k < KNN; ++k) { fd[k] = 3.0e38f; fi[k] = 0; }
#pragma unroll
        for (int src = 0; src < 2; ++src)
#pragma unroll
            for (int k = 0; k < KNN; ++k)
                topk_insert(fd, fi, s_d[wv][lane + 16 * src][k],
                                    s_i[wv][lane + 16 * src][k]);

        float tot = 0.0f, rs[KNN];
#pragma unroll
        for (int k = 0; k < KNN; ++k) { rs[k] = fast_rcp(fd[k] + 1e-8f); tot += rs[k]; }
        const float inv = fast_rcp(tot);
#pragma unroll
        for (int k = 0; k < KNN; ++k) { s_w[wv][lane][k] = rs[k] * inv; s_fi[wv][lane][k] = fi[k]; }
    }
    __syncthreads();

    // ---- gather + interpolate + merge, coalesced over C=256 ----------------
    const float* pcB  = pc  + (size_t)b * S * CCH;
    const float* pfB  = pf  + (size_t)b * N * CCH;
    float*       outB = out + (size_t)b * N * CCH;

    for (int p = 0; p < TPW; ++p) {
        const int n = n0 + p;
        float w[KNN]; int id[KNN];
#pragma unroll
        for (int k = 0; k < KNN; ++k) { w[k] = s_w[wv][p][k]; id[k] = s_fi[wv][p][k]; }

        for (int j = 0; j < CCH / WAVE; ++j) {
            const int c = lane + j * WAVE;
            float acc = 0.0f;
#pragma unroll
            for (int k = 0; k < KNN; ++k)
                acc = fmaf(w[k], pcB[(size_t)id[k] * CCH + c], acc);
            const float p1 = pfB[(size_t)n * CCH + c];
            outB[(size_t)n * CCH + c] =
                (fmaxf(p1, acc) + (p1 + acc) * 0.5f) * 0.5f;
        }
    }
}

// ---------------------------------------------------------------------------
// Host: three chained stages, intermediates in d_ws.
// Input order (setup_inputs dict): xyz0,nrm0,feat0, xyz1,nrm1,feat1,
//                                  xyz2,nrm2,feat2, xyz3,nrm3,feat3.
// Levels: 0:N=16384  1:4096  2:1024  3:256 (coarsest).  B=2, C=256.
// ---------------------------------------------------------------------------
extern "C" void kernel_launch(void* const* d_in, const int* in_sizes, int n_in,
                              void* d_out, int out_size, void* d_ws, size_t ws_size,
                              hipStream_t stream) {
    (void)in_sizes; (void)n_in; (void)out_size; (void)ws_size;

    const float* xyz0 = (const float*)d_in[0];
    const float* nrm0 = (const float*)d_in[1];
    const float* ft0  = (const float*)d_in[2];
    const float* xyz1 = (const float*)d_in[3];
    const float* nrm1 = (const float*)d_in[4];
    const float* ft1  = (const float*)d_in[5];
    const float* xyz2 = (const float*)d_in[6];
    const float* nrm2 = (const float*)d_in[7];
    const float* ft2  = (const float*)d_in[8];
    const float* xyz3 = (const float*)d_in[9];
    const float* nrm3 = (const float*)d_in[10];
    const float* ft3  = (const float*)d_in[11];

    float* x0 = (float*)d_ws;                                       // [2,1024,256]
    float* x1 = (float*)((char*)d_ws + (size_t)2 * 1024 * CCH * 4); // [2,4096,256]

    // Stage 0: fine=level2 (N=1024), coarse=level3 (S=256), features = feat3.
    propagate_kernel<<<dim3((2 * 1024) / (TPW * WPB)), BLOCK, 0, stream>>>(
        xyz2, xyz3, nrm2, nrm3, ft2, ft3, x0, 1024, 256);

    // Stage 1: fine=level1 (N=4096), coarse=level2 (S=1024), features = x0.
    propagate_kernel<<<dim3((2 * 4096) / (TPW * WPB)), BLOCK, 0, stream>>>(
        xyz1, xyz2, nrm1, nrm2, ft1, x0, x1, 4096, 1024);

    // Stage 2: fine=level0 (N=16384), coarse=level1 (S=4096), features = x1.
    propagate_kernel<<<dim3((2 * 16384) / (TPW * WPB)), BLOCK, 0, stream>>>(
        xyz0, xyz1, nrm0, nrm1, ft0, x1, (float*)d_out, 16384, 4096);
}